// HNHNLayer_2576980378148
// MI455X (gfx1250) — compile-verified
//
#include <hip/hip_runtime.h>
#include <hip/hip_bf16.h>

// ---------------------------------------------------------------------------
// HNHN layer for MI455X (gfx1250, wave32).
//   pass1: scatter nodes->hyperedges, fused GEMM  relu(X W_ve^T+b) + E W_ef^T+b
//   pass2: scatter hyperedges->nodes, GEMM        relu(X W_ev^T+b)
// GEMMs: V_WMMA_F32_16X16X4_F32 fed from LDS (coalesced b128 staging,
// bank-conflict-free ds_load_b64 operand reads).
// ---------------------------------------------------------------------------

#define N_NODES  100000
#define N_HEDGES 40000
#define N_INC    600000
#define DIM      128
#define LSTRIDE  132           // LDS row stride (floats): 128 + 4 pad
                               // -> read banks 4*m + kb mod 64, conflict-free

typedef float v2f __attribute__((ext_vector_type(2)));
typedef float v8f __attribute__((ext_vector_type(8)));

// -------------------------------------------------------------- zero scratch
__global__ void hnhn_zero_kernel(float4* __restrict__ p, int n4) {
    int i      = blockIdx.x * blockDim.x + threadIdx.x;
    int stride = gridDim.x * blockDim.x;
    float4 z = make_float4(0.f, 0.f, 0.f, 0.f);
    for (; i < n4; i += stride) p[i] = z;
}

// ---------------------------------------------------- weighted scatter-add
// One wave (32 lanes) per incidence pair; each lane handles 4 contiguous
// columns (coalesced float4 gather + 4 no-return global f32 atomics).
//   out[dst[i], :] += (num_w[src[i]] / den_w[dst[i]]) * feat[src[i], :]
__global__ __launch_bounds__(256) void hnhn_scatter_kernel(
    const float* __restrict__ feat,      // [Nsrc, 128]
    const int*   __restrict__ src_idx,   // [I]
    const int*   __restrict__ dst_idx,   // [I]
    const float* __restrict__ num_w,     // [Nsrc]
    const float* __restrict__ den_w,     // [Ndst]
    float*       __restrict__ out,       // [Ndst, 128] accumulators
    int I)
{
    int t    = blockIdx.x * blockDim.x + threadIdx.x;
    int i    = t >> 5;          // incidence index (one wave each)
    int lane = t & 31;          // column group: 4 floats per lane
    if (i >= I) return;

    int   s = src_idx[i];
    int   d = dst_idx[i];
    float w = num_w[s] / den_w[d];

    const float4* fv = (const float4*)(feat + (size_t)s * DIM);
    float4 v = fv[lane];

    float* o = out + (size_t)d * DIM + lane * 4;
    atomicAdd(o + 0, w * v.x);
    atomicAdd(o + 1, w * v.y);
    atomicAdd(o + 2, w * v.z);
    atomicAdd(o + 3, w * v.w);
}

// ------------------------------------------------------------- WMMA GEMM
// Y[R,128] = relu(X1 @ W1^T + b1)  [ + X2 @ W2^T + b2  if X2 != null ]
//
// Block = 64 threads = 2 waves; each wave computes a 16-row strip across all
// 128 output columns. The wave's X strip(s) are staged into LDS once with
// coalesced b128 loads, then WMMA operands come from bank-conflict-free
// ds_load_b64 reads (row stride LSTRIDE=132). W stays in global: it is 64 KB
// shared by every block -> L1/L2 resident.
//
// Wave32 fp32 WMMA VGPR layouts (ISA 7.12.2):
//   A (16x4):  lane l -> row m=l&15, K = 2*(l>>4)+{0,1}   (v2f)
//   B (4x16):  mirror; B[k][n] = W[n][k] -> same addressing with W row n
//   C/D:       reg r of lane l -> Y[row0 + r + 8*(l>>4)][nt*16 + (l&15)]
__global__ __launch_bounds__(64) void hnhn_gemm_kernel(
    const float* __restrict__ X1, const float* __restrict__ W1,
    const float* __restrict__ b1,
    const float* __restrict__ X2, const float* __restrict__ W2,
    const float* __restrict__ b2,
    float* __restrict__ Y, int R)
{
    __shared__ float lds[2 * 2 * 16 * LSTRIDE];   // 2 waves x 2 strips

    const int lane = threadIdx.x & 31;
    const int wave = threadIdx.x >> 5;            // 0..1
    const int row0 = (blockIdx.x * 2 + wave) * 16;
    const bool active = row0 < R;                  // wave-uniform
    const int row0c = active ? row0 : 0;           // clamp for safe staging

    float* s1 = lds + wave * (2 * 16 * LSTRIDE);   // X1 strip
    float* s2 = s1 + 16 * LSTRIDE;                 // X2 strip

    // ---- stage strips: coalesced 512B rows (global b128 -> ds b128) ------
    #pragma unroll 4
    for (int r = 0; r < 16; ++r) {
        float4 v = *(const float4*)(X1 + (size_t)(row0c + r) * DIM + lane * 4);
        *(float4*)(s1 + r * LSTRIDE + lane * 4) = v;
    }
    if (X2) {
        #pragma unroll 4
        for (int r = 0; r < 16; ++r) {
            float4 v = *(const float4*)(X2 + (size_t)(row0c + r) * DIM + lane * 4);
            *(float4*)(s2 + r * LSTRIDE + lane * 4) = v;
        }
    }
    __syncthreads();          // single barrier, above all divergence
    if (!active) return;

    const int m  = lane & 15;   // A row within strip == output column-in-tile
    const int kg = lane >> 4;   // lane half: K sub-pair selector
    const float* a1row = s1 + m * LSTRIDE;
    const float* a2row = s2 + m * LSTRIDE;

    for (int nt = 0; nt < 8; ++nt) {                  // 8 column tiles of 16
        const int    ncol = nt * 16 + m;
        const float* wrow = W1 + (size_t)ncol * DIM;

        v8f acc = {0.f, 0.f, 0.f, 0.f, 0.f, 0.f, 0.f, 0.f};
        #pragma unroll 4
        for (int kk = 0; kk < 32; ++kk) {             // K = 128, 4 per WMMA
            const int kb = kk * 4 + kg * 2;           // 8B-aligned (kb even)
            v2f a = *(const v2f*)(a1row + kb);        // ds_load_b64
            v2f b; b.x = wrow[kb]; b.y = wrow[kb + 1];
            acc = __builtin_amdgcn_wmma_f32_16x16x4_f32(
                false, a, false, b, (short)0, acc, false, false);
        }

        const float bias1 = b1[ncol];
        #pragma unroll
        for (int r = 0; r < 8; ++r) acc[r] = fmaxf(acc[r] + bias1, 0.f);

        if (X2) {                                     // fused second GEMM
            const float* w2row = W2 + (size_t)ncol * DIM;
            v8f acc2 = {0.f, 0.f, 0.f, 0.f, 0.f, 0.f, 0.f, 0.f};
            #pragma unroll 4
            for (int kk = 0; kk < 32; ++kk) {
                const int kb = kk * 4 + kg * 2;
                v2f a = *(const v2f*)(a2row + kb);    // ds_load_b64
                v2f b; b.x = w2row[kb]; b.y = w2row[kb + 1];
                acc2 = __builtin_amdgcn_wmma_f32_16x16x4_f32(
                    false, a, false, b, (short)0, acc2, false, false);
            }
            const float bias2 = b2[ncol];
            #pragma unroll
            for (int r = 0; r < 8; ++r) acc[r] += acc2[r] + bias2;
        }

        #pragma unroll
        for (int r = 0; r < 8; ++r)
            Y[(size_t)(row0 + r + 8 * kg) * DIM + ncol] = acc[r];
    }
}

// ---------------------------------------------------------------------------
extern "C" void kernel_launch(void* const* d_in, const int* in_sizes, int n_in,
                              void* d_out, int out_size, void* d_ws, size_t ws_size,
                              hipStream_t stream) {
    const float* vfeat        = (const float*)d_in[0];
    const float* efeat        = (const float*)d_in[1];
    const float* v_reg_weight = (const float*)d_in[2];
    const float* v_reg_sum    = (const float*)d_in[3];
    const float* e_reg_weight = (const float*)d_in[4];
    const float* e_reg_sum    = (const float*)d_in[5];
    const float* W_ve         = (const float*)d_in[6];
    const float* b_ve         = (const float*)d_in[7];
    const float* W_ev         = (const float*)d_in[8];
    const float* b_ev         = (const float*)d_in[9];
    const float* W_ef         = (const float*)d_in[10];
    const float* b_ef         = (const float*)d_in[11];
    const int*   node_idx     = (const int*)d_in[12];
    const int*   edge_idx     = (const int*)d_in[13];

    float* vfeat_out = (float*)d_out;                         // [N_NODES, 128]
    float* efeat_out = vfeat_out + (size_t)N_NODES * DIM;     // [N_HEDGES,128]
    float* norm      = (float*)d_ws;  // reused: max(N,E)*128 f32 = 51.2 MB

    const int scatter_blocks = (N_INC * 32) / 256;            // 75000

    // ---- pass 1: nodes -> hyperedges -------------------------------------
    hnhn_zero_kernel<<<4096, 256, 0, stream>>>(
        (float4*)norm, (N_HEDGES * DIM) / 4);
    hnhn_scatter_kernel<<<scatter_blocks, 256, 0, stream>>>(
        vfeat, node_idx, edge_idx, v_reg_weight, e_reg_sum, norm, N_INC);
    hnhn_gemm_kernel<<<(N_HEDGES + 31) / 32, 64, 0, stream>>>(
        norm, W_ve, b_ve, efeat, W_ef, b_ef, efeat_out, N_HEDGES);

    // ---- pass 2: hyperedges -> nodes -------------------------------------
    hnhn_zero_kernel<<<8192, 256, 0, stream>>>(
        (float4*)norm, (N_NODES * DIM) / 4);
    hnhn_scatter_kernel<<<scatter_blocks, 256, 0, stream>>>(
        efeat_out, edge_idx, node_idx, e_reg_weight, v_reg_sum, norm, N_INC);
    hnhn_gemm_kernel<<<(N_NODES + 31) / 32, 64, 0, stream>>>(
        norm, W_ev, b_ev, nullptr, nullptr, nullptr, vfeat_out, N_NODES);
}